// ALIGNNPyGChargeEarlySelective_74637941670352
// MI455X (gfx1250) — compile-verified
//
#include <hip/hip_runtime.h>
#include <hip/hip_bf16.h>

// ---------------- problem constants ----------------
#define NN   50000   // nodes
#define NE   800000  // edges
#define NG   32      // graphs
#define NA   92      // atom features
#define NB   41      // bond features
#define CHN  16      // charge feature dim
#define DD   64      // hidden dim
#define HH   128     // head dim
#define NL   3       // layers
#define NS   4096    // target sites

typedef __attribute__((ext_vector_type(2))) float v2f;
typedef __attribute__((ext_vector_type(8))) float v8f;
typedef __attribute__((ext_vector_type(4))) unsigned int u32x4;
typedef __attribute__((ext_vector_type(4))) int i32x4;
typedef __attribute__((ext_vector_type(8))) int i32x8;

#if defined(__has_builtin)
#if __has_builtin(__builtin_amdgcn_tensor_load_to_lds)
#define HAVE_TDM 1
#endif
#endif

// fp32 WMMA: D(16x16) = A(16x4) * B(4x16) + C   -> v_wmma_f32_16x16x4_f32
__device__ __forceinline__ v8f wmma4(v2f a, v2f b, v8f c) {
  return __builtin_amdgcn_wmma_f32_16x16x4_f32(
      /*neg_a=*/false, a, /*neg_b=*/false, b,
      /*c_mod=*/(short)0, c, /*reuse_a=*/false, /*reuse_b=*/false);
}

__device__ __forceinline__ float softplusf(float x) {
  return x > 20.f ? x : log1pf(__expf(x));
}
__device__ __forceinline__ void atomAddF(float* p, float v) {
  __hip_atomic_fetch_add(p, v, __ATOMIC_RELAXED, __HIP_MEMORY_SCOPE_AGENT);
}

// TDM DMA of a contiguous [words] x 4B tensor from global to LDS.
// D# built per CDNA5 ISA 8.3/8.4: group0 {count=1, lds_addr, global_addr, type=2},
// group1 {data_size=4B, tensor_dim0=stride=words, tile_dim0=words, tile_dim1=1}.
__device__ __forceinline__ void tdm_load_words(const void* gptr, void* lds, unsigned words) {
#if defined(HAVE_TDM)
  unsigned long long ga = (unsigned long long)(uintptr_t)gptr;
  unsigned lo = (unsigned)(uintptr_t)lds;           // low 32 bits of generic = LDS byte addr
  u32x4 g0;
  g0.x = 1u;                                        // count=1 (user descriptor)
  g0.y = lo;                                        // lds_addr
  g0.z = (unsigned)(ga & 0xffffffffu);              // global_addr[31:0]
  g0.w = (unsigned)((ga >> 32) & 0x01ffffffu) | 0x80000000u; // addr[56:32] | type=2
  i32x8 g1;
  g1[0] = 0x20000;                  // data_size=2 -> 4 bytes
  g1[1] = (int)(words << 16);       // tensor_dim0[15:0] at bits 63:48
  g1[2] = (int)((words >> 16) & 0xffffu) | (1 << 16); // tensor_dim0 hi | tensor_dim1=1
  g1[3] = (int)(words << 16);       // tile_dim0 at bits 127:112
  g1[4] = 1;                        // tile_dim1=1, tile_dim2=0
  g1[5] = (int)words;               // tensor_dim0_stride lo32
  g1[6] = 0;
  g1[7] = 0;
  i32x4 gz = {0, 0, 0, 0};
#if __has_include(<hip/amd_detail/amd_gfx1250_TDM.h>)
  i32x8 gz8 = {0, 0, 0, 0, 0, 0, 0, 0};
  __builtin_amdgcn_tensor_load_to_lds(g0, g1, gz, gz, gz8, 0);
#else
  __builtin_amdgcn_tensor_load_to_lds(g0, g1, gz, gz, 0);
#endif
#if __has_builtin(__builtin_amdgcn_s_wait_tensorcnt)
  __builtin_amdgcn_s_wait_tensorcnt(0);
#else
  asm volatile("s_wait_tensorcnt 0x0" ::: "memory");
#endif
#else
  (void)gptr; (void)lds; (void)words;
#endif
}

// ---------------- tiny kernels ----------------
__global__ void k_charge(const float* __restrict__ charge, const float* __restrict__ Wc,
                         const float* __restrict__ bc, float* __restrict__ cf) {
  int i = threadIdx.x;               // 512 threads: 32 graphs x 16 channels
  int g = i >> 4, c = i & 15;
  cf[i] = charge[g] * Wc[c] + bc[c];
}

__global__ void k_zero4(float* __restrict__ p, long n4) {
  long i = (long)blockIdx.x * blockDim.x + threadIdx.x;
  if (i < n4) ((float4*)p)[i] = make_float4(0.f, 0.f, 0.f, 0.f);
}

// ---------------- atom embedding: h = [x | cf[batch]] @ Wa^T + ba  (K=108) ----------------
__global__ __launch_bounds__(32)
void k_node_embed(const float* __restrict__ x, const float* __restrict__ cf,
                  const int* __restrict__ batch,
                  const float* __restrict__ Wa, const float* __restrict__ ba,
                  float* __restrict__ h) {
  __shared__ float At[16 * 108];
  const int m0 = blockIdx.x * 16;
  const int tid = threadIdx.x;       // 32 threads (one wave)
  for (int i = tid; i < 16 * 108; i += 32) {
    int r = i / 108, c = i - r * 108;
    int node = m0 + r;
    At[i] = (c < NA) ? x[(size_t)node * NA + c] : cf[batch[node] * CHN + (c - NA)];
  }
  __syncthreads();
  const int lane = tid & 31, rsel = lane & 15, hi = lane >> 4;
#pragma unroll
  for (int ct = 0; ct < 4; ++ct) {
    v8f acc = {};
#pragma unroll
    for (int kk = 0; kk < 27; ++kk) {          // 27*4 = 108 exactly
      int k = kk * 4 + (hi << 1);
      v2f a; a.x = At[rsel * 108 + k]; a.y = At[rsel * 108 + k + 1];
      const float* qw = Wa + (size_t)(ct * 16 + rsel) * 108 + k;
      v2f b; b.x = qw[0]; b.y = qw[1];
      acc = wmma4(a, b, acc);
    }
    const int n = ct * 16 + rsel;
    const float bv = ba[n];
    const int rb = m0 + (hi << 3);
#pragma unroll
    for (int r = 0; r < 8; ++r)
      h[(size_t)(rb + r) * DD + n] = acc[r] + bv;
  }
}

// ---------------- bond embedding: e = edge_attr @ Wb^T + bb  (K=41 -> pad 44) ----------------
__global__ __launch_bounds__(256)
void k_bond_embed(const float* __restrict__ ea, const float* __restrict__ Wb,
                  const float* __restrict__ bb, float* __restrict__ e) {
  __shared__ float Wt[64 * 44];
  __shared__ float At[8 * 16 * 44];
  const int tid = threadIdx.x;       // 256 threads = 8 waves
  for (int i = tid; i < 64 * 44; i += 256) {
    int o = i / 44, k = i - o * 44;
    Wt[i] = (k < NB) ? Wb[(size_t)o * NB + k] : 0.f;
  }
  const int wave = tid >> 5, lane = tid & 31;
  const int m0 = blockIdx.x * 128 + wave * 16;
  float* Aw = At + wave * 16 * 44;
  for (int i = lane; i < 16 * 44; i += 32) {
    int r = i / 44, k = i - r * 44;
    Aw[i] = (k < NB) ? ea[(size_t)(m0 + r) * NB + k] : 0.f;
  }
  __syncthreads();
  const int rsel = lane & 15, hi = lane >> 4;
#pragma unroll
  for (int ct = 0; ct < 4; ++ct) {
    v8f acc = {};
#pragma unroll
    for (int kk = 0; kk < 11; ++kk) {          // 11*4 = 44 (padded)
      int k = kk * 4 + (hi << 1);
      v2f a; a.x = Aw[rsel * 44 + k]; a.y = Aw[rsel * 44 + k + 1];
      v2f b; b.x = Wt[(ct * 16 + rsel) * 44 + k]; b.y = Wt[(ct * 16 + rsel) * 44 + k + 1];
      acc = wmma4(a, b, acc);
    }
    const int n = ct * 16 + rsel;
    const float bv = bb[n];
    const int rb = m0 + (hi << 3);
#pragma unroll
    for (int r = 0; r < 8; ++r)
      e[(size_t)(rb + r) * DD + n] = acc[r] + bv;
  }
}

// ---------------- generic node GEMM: Y[NN,64] = X[NN,64] @ W[64,64]^T + bias ----------------
__global__ __launch_bounds__(32)
void k_gemm64(const float* __restrict__ X, const float* __restrict__ W,
              const float* __restrict__ bias, float* __restrict__ Y) {
  const int m0 = blockIdx.x * 16;
  const int lane = threadIdx.x & 31, rsel = lane & 15, hi = lane >> 4;
  v2f a[16];
#pragma unroll
  for (int kk = 0; kk < 16; ++kk) {
    const float* q = X + (size_t)(m0 + rsel) * DD + kk * 4 + (hi << 1);
    a[kk].x = q[0]; a[kk].y = q[1];
  }
#pragma unroll
  for (int ct = 0; ct < 4; ++ct) {
    v8f acc = {};
#pragma unroll
    for (int kk = 0; kk < 16; ++kk) {
      const float* qw = W + (size_t)(ct * 16 + rsel) * DD + kk * 4 + (hi << 1);
      v2f b; b.x = qw[0]; b.y = qw[1];
      acc = wmma4(a[kk], b, acc);
    }
    const int n = ct * 16 + rsel;
    const float bv = bias[n];
    const int rb = m0 + (hi << 3);
#pragma unroll
    for (int r = 0; r < 8; ++r)
      Y[(size_t)(rb + r) * DD + n] = acc[r] + bv;
  }
}

// ---------------- fused edge kernel: TDM weight stage + WMMA GEMM + gates + BN + scatter ----
__global__ __launch_bounds__(256)
void k_edge_fused(const float* __restrict__ hs, const float* __restrict__ hd,
                  const float* __restrict__ hu, float* __restrict__ e,
                  const int* __restrict__ ei,     // [2, NE]
                  const float* __restrict__ Wg, const float* __restrict__ bg,
                  const float* __restrict__ bnp,  // [4,64] gamma,beta,mean,var
                  float* __restrict__ hsum, float* __restrict__ ssum) {
  __shared__ float Wt[64 * 64];
  const int tid = threadIdx.x;       // 256 threads = 8 waves, 128 edges/block
#if defined(HAVE_TDM)
  if ((tid >> 5) == 0) {             // one wave issues the DMA and waits on TENSORcnt
    tdm_load_words(Wg, Wt, 64 * 64);
  }
#else
  for (int i = tid; i < 4096; i += 256) Wt[i] = Wg[i];
#endif
  __syncthreads();

  const int wave = tid >> 5, lane = tid & 31;
  const int m0 = blockIdx.x * 128 + wave * 16;
  const int rsel = lane & 15, hi = lane >> 4;

  // preload this wave's A fragments (16 edge rows of e)
  v2f a[16];
#pragma unroll
  for (int kk = 0; kk < 16; ++kk) {
    const float* q = e + (size_t)(m0 + rsel) * DD + kk * 4 + (hi << 1);
    a[kk].x = q[0]; a[kk].y = q[1];
  }

  // per-lane src/dst indices for the 8 output rows it owns
  const int rb = m0 + (hi << 3);
  int rs[8], cd[8];
#pragma unroll
  for (int r = 0; r < 8; ++r) { rs[r] = ei[rb + r]; cd[r] = ei[NE + rb + r]; }
  __builtin_prefetch(&hs[(size_t)rs[0] * DD], 0, 0);   // global_prefetch_b8
  __builtin_prefetch(&hd[(size_t)cd[0] * DD], 0, 0);

#pragma unroll
  for (int ct = 0; ct < 4; ++ct) {
    v8f acc = {};
#pragma unroll
    for (int kk = 0; kk < 16; ++kk) {
      int k = kk * 4 + (hi << 1);
      v2f b; b.x = Wt[(ct * 16 + rsel) * 64 + k]; b.y = Wt[(ct * 16 + rsel) * 64 + k + 1];
      acc = wmma4(a[kk], b, acc);
    }
    const int n = ct * 16 + rsel;
    const float be = bg[n];
    const float gam = bnp[n], bet = bnp[64 + n], mu = bnp[128 + n], var = bnp[192 + n];
    const float sc = rsqrtf(var + 1e-5f) * gam;
#pragma unroll
    for (int r = 0; r < 8; ++r) {
      const int eid = rb + r;
      const float mval = acc[r] + be + hs[(size_t)rs[r] * DD + n] + hd[(size_t)cd[r] * DD + n];
      const float sig = 1.f / (1.f + __expf(-mval));
      const float sp = softplusf((mval - mu) * sc + bet);
      e[(size_t)eid * DD + n] += sp;                       // edge residual
      atomAddF(&hsum[(size_t)cd[r] * DD + n], hu[(size_t)cd[r] * DD + n] * sig);
      atomAddF(&ssum[(size_t)cd[r] * DD + n], sig);
    }
  }
}

// ---------------- node update (elementwise) ----------------
__global__ void k_node_update(float* __restrict__ h, const float* __restrict__ hv,
                              const float* __restrict__ hsum, const float* __restrict__ ssum,
                              const float* __restrict__ bn_n, const float* __restrict__ bn_o) {
  const long i = (long)blockIdx.x * blockDim.x + threadIdx.x;
  if (i >= (long)NN * DD) return;
  const int n = (int)(i & 63);
  const float x1 = hv[i] + hsum[i] / (ssum[i] + 1e-6f);
  const float b1 = (x1 - bn_n[128 + n]) * rsqrtf(bn_n[192 + n] + 1e-5f) * bn_n[n] + bn_n[64 + n];
  const float h0 = h[i];
  const float x2 = h0 + softplusf(b1);
  const float b2 = (x2 - bn_o[128 + n]) * rsqrtf(bn_o[192 + n] + 1e-5f) * bn_o[n] + bn_o[64 + n];
  h[i] = softplusf(b2) + h0;
}

// ---------------- selective pooling ----------------
__global__ void k_pool(const float* __restrict__ h, const int* __restrict__ tsi,
                       const int* __restrict__ batch, float* __restrict__ sums,
                       float* __restrict__ cnt) {
  const int i = blockIdx.x * 256 + threadIdx.x;   // NS*DD = 262144 exact
  const int s = i >> 6, d = i & 63;
  const int nd = tsi[s];
  const int b = batch[nd];
  atomAddF(&sums[b * 64 + d], h[(size_t)nd * 64 + d]);
  if (d == 0) atomAddF(&cnt[b], 1.f);
}

// ---------------- prediction head (tiny) ----------------
__global__ void k_head(const float* __restrict__ sums, const float* __restrict__ cnt,
                       const float* __restrict__ W1, const float* __restrict__ b1,
                       const float* __restrict__ W2, const float* __restrict__ b2,
                       const float* __restrict__ W3, const float* __restrict__ b3,
                       float* __restrict__ out) {
  __shared__ float rep[NG * DD];
  __shared__ float z1[NG * HH];
  __shared__ float z2[NG * HH];
  const int tid = threadIdx.x;      // 128
  for (int i = tid; i < NG * DD; i += 128) {
    int g = i >> 6;
    float c = cnt[g];
    rep[i] = (c > 0.f) ? sums[i] / fmaxf(c, 1.f) : 0.f;
  }
  __syncthreads();
  for (int g = 0; g < NG; ++g) {
    float acc = b1[tid];
    for (int d = 0; d < DD; ++d) acc += rep[g * DD + d] * W1[tid * DD + d];
    z1[g * HH + tid] = softplusf(acc);
  }
  __syncthreads();
  for (int g = 0; g < NG; ++g) {
    float acc = b2[tid];
    for (int d = 0; d < HH; ++d) acc += z1[g * HH + d] * W2[tid * HH + d];
    z2[g * HH + tid] = softplusf(acc);
  }
  __syncthreads();
  if (tid < NG) {
    float acc = b3[0];
    for (int d = 0; d < HH; ++d) acc += z2[tid * HH + d] * W3[d];
    out[tid] = acc;
  }
}

// ---------------- host launcher ----------------
static inline size_t alup(size_t x) { return (x + 255) & ~(size_t)255; }

extern "C" void kernel_launch(void* const* d_in, const int* in_sizes, int n_in,
                              void* d_out, int out_size, void* d_ws, size_t ws_size,
                              hipStream_t stream) {
  const float* x          = (const float*)d_in[0];
  const float* edge_attr  = (const float*)d_in[1];
  const float* charge     = (const float*)d_in[2];
  const float* W_charge   = (const float*)d_in[3];
  const float* b_charge   = (const float*)d_in[4];
  const float* W_atom     = (const float*)d_in[5];
  const float* b_atom     = (const float*)d_in[6];
  const float* W_bond     = (const float*)d_in[7];
  const float* b_bond     = (const float*)d_in[8];
  const float* src_gate_W = (const float*)d_in[9];
  const float* src_gate_b = (const float*)d_in[10];
  const float* dst_gate_W = (const float*)d_in[11];
  const float* dst_gate_b = (const float*)d_in[12];
  const float* edge_gate_W= (const float*)d_in[13];
  const float* edge_gate_b= (const float*)d_in[14];
  const float* src_upd_W  = (const float*)d_in[15];
  const float* src_upd_b  = (const float*)d_in[16];
  const float* dst_upd_W  = (const float*)d_in[17];
  const float* dst_upd_b  = (const float*)d_in[18];
  const float* bn_edges   = (const float*)d_in[19];
  const float* bn_nodes   = (const float*)d_in[20];
  const float* bn_outer   = (const float*)d_in[21];
  const float* pred_W1    = (const float*)d_in[22];
  const float* pred_b1    = (const float*)d_in[23];
  const float* pred_W2    = (const float*)d_in[24];
  const float* pred_b2    = (const float*)d_in[25];
  const float* pred_W3    = (const float*)d_in[26];
  const float* pred_b3    = (const float*)d_in[27];
  const int*   edge_index = (const int*)d_in[28];
  const int*   batch      = (const int*)d_in[29];
  const int*   tsi        = (const int*)d_in[30];
  float* out = (float*)d_out;

  // workspace carve-up
  char* ws = (char*)d_ws;
  size_t off = 0;
  float* cf    = (float*)(ws + off); off = alup(off + (size_t)NG * CHN * 4);
  float* h     = (float*)(ws + off); off = alup(off + (size_t)NN * DD * 4);
  float* e     = (float*)(ws + off); off = alup(off + (size_t)NE * DD * 4);
  float* hs    = (float*)(ws + off); off = alup(off + (size_t)NN * DD * 4);
  float* hd    = (float*)(ws + off); off = alup(off + (size_t)NN * DD * 4);
  float* hu    = (float*)(ws + off); off = alup(off + (size_t)NN * DD * 4);
  float* hv    = (float*)(ws + off); off = alup(off + (size_t)NN * DD * 4);
  float* hsum  = (float*)(ws + off); off = alup(off + (size_t)NN * DD * 4);
  float* ssum  = (float*)(ws + off); off = alup(off + (size_t)NN * DD * 4);
  float* psum  = (float*)(ws + off); off = alup(off + (size_t)NG * DD * 4);
  float* pcnt  = (float*)(ws + off); off = alup(off + (size_t)NG * 4);
  (void)ws_size; (void)n_in; (void)in_sizes; (void)out_size;

  // embeddings
  k_charge<<<1, NG * CHN, 0, stream>>>(charge, W_charge, b_charge, cf);
  k_node_embed<<<NN / 16, 32, 0, stream>>>(x, cf, batch, W_atom, b_atom, h);
  k_bond_embed<<<NE / 128, 256, 0, stream>>>(edge_attr, W_bond, b_bond, e);

  for (int l = 0; l < NL; ++l) {
    const size_t wo = (size_t)l * DD * DD, bo = (size_t)l * DD, no = (size_t)l * 4 * DD;
    k_gemm64<<<NN / 16, 32, 0, stream>>>(h, src_gate_W + wo, src_gate_b + bo, hs);
    k_gemm64<<<NN / 16, 32, 0, stream>>>(h, dst_gate_W + wo, dst_gate_b + bo, hd);
    k_gemm64<<<NN / 16, 32, 0, stream>>>(h, dst_upd_W + wo, dst_upd_b + bo, hu);
    k_gemm64<<<NN / 16, 32, 0, stream>>>(h, src_upd_W + wo, src_upd_b + bo, hv);
    // zero hsum + ssum (contiguous 2*NN*DD floats)
    {
      long n4 = (long)NN * DD * 2 / 4;
      k_zero4<<<(unsigned)((n4 + 255) / 256), 256, 0, stream>>>(hsum, n4);
    }
    k_edge_fused<<<NE / 128, 256, 0, stream>>>(hs, hd, hu, e, edge_index,
                                               edge_gate_W + wo, edge_gate_b + bo,
                                               bn_edges + no, hsum, ssum);
    k_node_update<<<(NN * DD) / 256, 256, 0, stream>>>(h, hv, hsum, ssum,
                                                       bn_nodes + no, bn_outer + no);
  }

  // pooling + head
  {
    long n4 = ((long)NG * DD + NG) / 4;   // 2080 floats, contiguous psum|pcnt
    k_zero4<<<(unsigned)((n4 + 255) / 256), 256, 0, stream>>>(psum, n4);
  }
  k_pool<<<(NS * DD) / 256, 256, 0, stream>>>(h, tsi, batch, psum, pcnt);
  k_head<<<1, HH, 0, stream>>>(psum, pcnt, pred_W1, pred_b1, pred_W2, pred_b2,
                               pred_W3, pred_b3, out);
}